// DaGCN_84267258347721
// MI455X (gfx1250) — compile-verified
//
#include <hip/hip_runtime.h>
#include <hip/hip_bf16.h>
#include <math.h>

typedef float v2f __attribute__((ext_vector_type(2)));
typedef float v8f __attribute__((ext_vector_type(8)));

#define NFEAT 256
#define NHID  64
#define NCLASS 32

// ---------------------------------------------------------------------------
// Zero-fill (workspace init each call; harness does not re-zero between runs)
// ---------------------------------------------------------------------------
__global__ void zero_f32(float* __restrict__ p, long n) {
    long i = (long)blockIdx.x * blockDim.x + threadIdx.x;
    if (i < n) p[i] = 0.0f;
}

// ---------------------------------------------------------------------------
// Dense GEMM  S[Nrows,Ncols] = X[Nrows,K] @ W[K,Ncols]  via V_WMMA_F32_16X16X4_F32
// One wave32 per 16-row x Ncols output strip: CT = Ncols/16 independent
// accumulators -> CT parallel WMMA chains per K-step (no serial RAW through a
// single accumulator) and the A fragment is loaded once per K-step and reused
// for all CT column tiles.
//
// A-fragment (16x4 f32, ISA 7.12.2): lanes 0-15 hold row M=lane, K={0,1} in
// v0/v1; lanes 16-31 hold K={2,3}.  -> contiguous float2 load per lane.
// B-fragment (4x16): mirrored (lane group = K-half, vgpr = K parity).
// C/D (16x16 f32): vgpr r -> M=r (lanes 0-15), M=r+8 (lanes 16-31), N=lane&15.
// ---------------------------------------------------------------------------
template <int CT>
__global__ __launch_bounds__(256)
void wmma_gemm_f32(const float* __restrict__ X,
                   const float* __restrict__ W,
                   float* __restrict__ S,
                   int Nrows, int K) {
    const int Ncols = CT * 16;
    const int wave = blockIdx.x * (blockDim.x >> 5) + (threadIdx.x >> 5);
    const int row0 = wave * 16;
    if (row0 >= Nrows) return;               // wave-uniform exit (EXEC all-1 for WMMA)

    const int lane = threadIdx.x & 31;
    const int lr   = lane & 15;              // A row / B,C column within tile
    const int kh   = lane >> 4;              // K-half selector (0 or 1)

    const float* __restrict__ xp = X + (size_t)(row0 + lr) * K + (kh << 1);
    const float* __restrict__ wp = W + lr;

    const v8f vzero = {};
    v8f acc[CT];
#pragma unroll
    for (int t = 0; t < CT; ++t) acc[t] = vzero;

    for (int k0 = 0; k0 < K; k0 += 4) {
        const v2f a = *(const v2f*)(xp + k0);       // X[row][k0+2*kh .. +1]
        const int kb = k0 + (kh << 1);
#pragma unroll
        for (int t = 0; t < CT; ++t) {
            v2f b;
            b.x = wp[(size_t)kb * Ncols + t * 16];        // W[kb  ][tile t col]
            b.y = wp[(size_t)(kb + 1) * Ncols + t * 16];  // W[kb+1][tile t col]
            acc[t] = __builtin_amdgcn_wmma_f32_16x16x4_f32(
                /*neg_a=*/false, a, /*neg_b=*/false, b,
                /*c_mod=*/(short)0, acc[t], /*reuse_a=*/false, /*reuse_b=*/false);
        }
    }

#pragma unroll
    for (int t = 0; t < CT; ++t) {
        float* __restrict__ op =
            S + (size_t)(row0 + kh * 8) * Ncols + t * 16 + lr;
#pragma unroll
        for (int r = 0; r < 8; ++r)
            op[(size_t)r * Ncols] = acc[t][r];
    }
}

// ---------------------------------------------------------------------------
// Edge scatter-add:  AGG[dst[e], colOff + f] += ew[e] * S[src[e], f]
// One thread per (edge, 4 features): float4 gather + 4 global_atomic_add_f32.
// AGG buffers are L2-resident (<=26MB vs 192MB L2) so RMWs stay on-chip.
// ---------------------------------------------------------------------------
__global__ void edge_scatter4(const float* __restrict__ S, int ldS,
                              const int* __restrict__ src,
                              const int* __restrict__ dst,
                              const float* __restrict__ ew,
                              int E, int Dq,               // Dq = D/4
                              float* __restrict__ AGG, int ldA, int colOff) {
    const int tid = blockIdx.x * blockDim.x + threadIdx.x;
    const int e = tid / Dq;
    if (e >= E) return;
    const int fq = (tid - e * Dq) << 2;
    const int sN = src[e];
    const int dN = dst[e];
    const float w = ew[e];
    const float4 v = *(const float4*)(S + (size_t)sN * ldS + fq);
    float* ap = AGG + (size_t)dN * ldA + colOff + fq;
    atomicAdd(ap + 0, w * v.x);
    atomicAdd(ap + 1, w * v.y);
    atomicAdd(ap + 2, w * v.z);
    atomicAdd(ap + 3, w * v.w);
}

// ---------------------------------------------------------------------------
// X[N,128] : X[:,0:64] += b0, X[:,64:128] += b1, then ReLU, in place.
// ---------------------------------------------------------------------------
__global__ void bias_relu128(float* __restrict__ X,
                             const float* __restrict__ b0,
                             const float* __restrict__ b1, long n) {
    long i = (long)blockIdx.x * blockDim.x + threadIdx.x;
    if (i >= n) return;
    const int c = (int)(i & 127);
    const float bv = (c < 64) ? b0[c] : b1[c - 64];
    const float v = X[i] + bv;
    X[i] = v > 0.0f ? v : 0.0f;
}

__device__ __forceinline__ float sigmoidf(float x) {
    return 1.0f / (1.0f + __expf(-x));
}

// ---------------------------------------------------------------------------
// Layer-1 gating: one wave32 per node, 4 features per lane (128 total).
// lam_i = sigmoid(x_i . g_iw + g_ib);  w = [lam1,lam2]/max(lam1+lam2,1e-12)
// X1 <- w0*X1 + w1*X2  (in place)
// ---------------------------------------------------------------------------
__global__ void gate_combine(float* __restrict__ X1,
                             const float* __restrict__ X2,
                             const float* __restrict__ g1w, const float* __restrict__ g1b,
                             const float* __restrict__ g2w, const float* __restrict__ g2b,
                             int N) {
    const int node = blockIdx.x * (blockDim.x >> 5) + (threadIdx.x >> 5);
    if (node >= N) return;
    const int lane = threadIdx.x & 31;
    const size_t base = (size_t)node * 128 + (size_t)lane * 4;

    const float4 a  = *(const float4*)(X1 + base);
    const float4 b  = *(const float4*)(X2 + base);
    const float4 w1 = *(const float4*)(g1w + lane * 4);
    const float4 w2 = *(const float4*)(g2w + lane * 4);

    float d1 = a.x * w1.x + a.y * w1.y + a.z * w1.z + a.w * w1.w;
    float d2 = b.x * w2.x + b.y * w2.y + b.z * w2.z + b.w * w2.w;
#pragma unroll
    for (int m = 16; m > 0; m >>= 1) {
        d1 += __shfl_xor(d1, m);
        d2 += __shfl_xor(d2, m);
    }
    const float lam1 = sigmoidf(d1 + g1b[0]);
    const float lam2 = sigmoidf(d2 + g2b[0]);
    const float inv  = 1.0f / fmaxf(lam1 + lam2, 1e-12f);
    const float c0 = lam1 * inv, c1 = lam2 * inv;

    float4 o;
    o.x = c0 * a.x + c1 * b.x;
    o.y = c0 * a.y + c1 * b.y;
    o.z = c0 * a.z + c1 * b.z;
    o.w = c0 * a.w + c1 * b.w;
    *(float4*)(X1 + base) = o;
}

// ---------------------------------------------------------------------------
// Layer-2 gating + output packing: one wave32 per node, 1 feature per lane.
// p_i = agg_i + b2;  mu_i = sigmoid(p_i . h_iw + h_ib);  L1-normalize;
// d_out = [ out | gnn_prop1 | gnn_prop2 ] flat.
// ---------------------------------------------------------------------------
__global__ void final_gate(const float* __restrict__ P1,
                           const float* __restrict__ P2,
                           const float* __restrict__ b2,
                           const float* __restrict__ h1w, const float* __restrict__ h1b,
                           const float* __restrict__ h2w, const float* __restrict__ h2b,
                           float* __restrict__ out, int N) {
    const int node = blockIdx.x * (blockDim.x >> 5) + (threadIdx.x >> 5);
    if (node >= N) return;
    const int lane = threadIdx.x & 31;
    const size_t idx = (size_t)node * NCLASS + lane;

    const float bb = b2[lane];
    const float p1 = P1[idx] + bb;
    const float p2 = P2[idx] + bb;

    float d1 = p1 * h1w[lane];
    float d2 = p2 * h2w[lane];
#pragma unroll
    for (int m = 16; m > 0; m >>= 1) {
        d1 += __shfl_xor(d1, m);
        d2 += __shfl_xor(d2, m);
    }
    const float mu1 = sigmoidf(d1 + h1b[0]);
    const float mu2 = sigmoidf(d2 + h2b[0]);
    const float inv = 1.0f / fmaxf(mu1 + mu2, 1e-12f);

    out[idx]                            = (mu1 * p1 + mu2 * p2) * inv;
    out[(size_t)N * NCLASS + idx]       = p1;
    out[(size_t)2 * N * NCLASS + idx]   = p2;
}

// ---------------------------------------------------------------------------
extern "C" void kernel_launch(void* const* d_in, const int* in_sizes, int n_in,
                              void* d_out, int out_size, void* d_ws, size_t ws_size,
                              hipStream_t stream) {
    const float* x1a = (const float*)d_in[0];
    const float* x1b = (const float*)d_in[1];
    const float* x2a = (const float*)d_in[2];
    const float* x2b = (const float*)d_in[3];
    const int*   src1 = (const int*)d_in[4];
    const int*   dst1 = (const int*)d_in[5];
    const float* ew1  = (const float*)d_in[6];
    const int*   src2 = (const int*)d_in[7];
    const int*   dst2 = (const int*)d_in[8];
    const float* ew2  = (const float*)d_in[9];
    const float* W1a = (const float*)d_in[10];
    const float* b1a = (const float*)d_in[11];
    const float* W1b = (const float*)d_in[12];
    const float* b1b = (const float*)d_in[13];
    const float* W2  = (const float*)d_in[14];
    const float* b2  = (const float*)d_in[15];
    const float* g1w = (const float*)d_in[16];
    const float* g1b = (const float*)d_in[17];
    const float* g2w = (const float*)d_in[18];
    const float* g2b = (const float*)d_in[19];
    const float* h1w = (const float*)d_in[20];
    const float* h1b = (const float*)d_in[21];
    const float* h2w = (const float*)d_in[22];
    const float* h2b = (const float*)d_in[23];

    const int N = in_sizes[0] / NFEAT;   // 50000
    const int E = in_sizes[4];           // 800000

    float* x1buf = (float*)d_ws;                       // [N,128]
    float* x2buf = x1buf + (size_t)N * 128;            // [N,128]
    float* sbuf  = x2buf + (size_t)N * 128;            // [N,64]  (reused)
    float* s2buf = sbuf  + (size_t)N * 64;             // [N,32]
    float* p1    = s2buf + (size_t)N * 32;             // [N,32]
    float* p2    = p1    + (size_t)N * 32;             // [N,32]

    // Zero the accumulation targets (x1buf+x2buf contiguous; p1+p2 contiguous)
    {
        long nz = (long)N * 256;
        zero_f32<<<(int)((nz + 255) / 256), 256, 0, stream>>>(x1buf, nz);
        long nz2 = (long)N * 64;
        zero_f32<<<(int)((nz2 + 255) / 256), 256, 0, stream>>>(p1, nz2);
    }

    // Layer 1: four GEMM -> scatter pairs (sbuf reused; stream serializes)
    const int waves1  = N / 16;                          // one wave per 16-row strip
    const int blocks1 = (waves1 + 7) / 8;                // 8 waves / 256-thread block
    const long scat1  = (long)E * (NHID / 4);
    const int sblk1   = (int)((scat1 + 255) / 256);

    wmma_gemm_f32<4><<<blocks1, 256, 0, stream>>>(x1a, W1a, sbuf, N, NFEAT);
    edge_scatter4<<<sblk1, 256, 0, stream>>>(sbuf, NHID, src1, dst1, ew1, E, NHID / 4,
                                             x1buf, 128, 0);
    wmma_gemm_f32<4><<<blocks1, 256, 0, stream>>>(x1b, W1b, sbuf, N, NFEAT);
    edge_scatter4<<<sblk1, 256, 0, stream>>>(sbuf, NHID, src1, dst1, ew1, E, NHID / 4,
                                             x1buf, 128, 64);
    wmma_gemm_f32<4><<<blocks1, 256, 0, stream>>>(x2a, W1a, sbuf, N, NFEAT);
    edge_scatter4<<<sblk1, 256, 0, stream>>>(sbuf, NHID, src2, dst2, ew2, E, NHID / 4,
                                             x2buf, 128, 0);
    wmma_gemm_f32<4><<<blocks1, 256, 0, stream>>>(x2b, W1b, sbuf, N, NFEAT);
    edge_scatter4<<<sblk1, 256, 0, stream>>>(sbuf, NHID, src2, dst2, ew2, E, NHID / 4,
                                             x2buf, 128, 64);

    // Bias + ReLU on both views
    {
        long n = (long)N * 128;
        int blk = (int)((n + 255) / 256);
        bias_relu128<<<blk, 256, 0, stream>>>(x1buf, b1a, b1b, n);
        bias_relu128<<<blk, 256, 0, stream>>>(x2buf, b1a, b1b, n);
    }

    // Gated combine (in place into x1buf)
    {
        int blk = (N + 7) / 8;                            // 8 nodes per 256-thread block
        gate_combine<<<blk, 256, 0, stream>>>(x1buf, x2buf, g1w, g1b, g2w, g2b, N);
    }

    // Layer 2: GEMM [N,128]@[128,32] then two scatters
    {
        wmma_gemm_f32<2><<<blocks1, 256, 0, stream>>>(x1buf, W2, s2buf, N, 2 * NHID);

        const long scat2 = (long)E * (NCLASS / 4);
        const int sblk2  = (int)((scat2 + 255) / 256);
        edge_scatter4<<<sblk2, 256, 0, stream>>>(s2buf, NCLASS, src1, dst1, ew1, E,
                                                 NCLASS / 4, p1, NCLASS, 0);
        edge_scatter4<<<sblk2, 256, 0, stream>>>(s2buf, NCLASS, src2, dst2, ew2, E,
                                                 NCLASS / 4, p2, NCLASS, 0);
    }

    // Final gating + pack outputs
    {
        int blk = (N + 7) / 8;
        final_gate<<<blk, 256, 0, stream>>>(p1, p2, b2, h1w, h1b, h2w, h2b,
                                            (float*)d_out, N);
    }
}